// TensorForceNet_58016418234653
// MI455X (gfx1250) — compile-verified
//
#include <hip/hip_runtime.h>
#include <hip/hip_bf16.h>
#include <math.h>

typedef float v2f __attribute__((ext_vector_type(2)));
typedef float v8f __attribute__((ext_vector_type(8)));

#define HD 128
#define CUTOFF 4.5f

// ---------------------------------------------------------------- utilities
__global__ void k_zero(float* __restrict__ p, size_t n) {
    size_t i = (size_t)blockIdx.x * blockDim.x + threadIdx.x;
    if (i < n) p[i] = 0.0f;
}

// Zemb[n,h] = emb_w[z[n], h]
__global__ void k_embed(float* __restrict__ Zemb, const float* __restrict__ emb_w,
                        const int* __restrict__ z, int Nn) {
    size_t i = (size_t)blockIdx.x * blockDim.x + threadIdx.x;
    if (i >= (size_t)Nn * HD) return;
    int n = (int)(i >> 7), h = (int)(i & 127);
    Zemb[i] = emb_w[(size_t)z[n] * HD + h];
}

// per-edge geometry: rbf(32), normalized vec, cosine cutoff
__global__ void k_edge_geom(const float* __restrict__ pos, const int* __restrict__ ei, int E,
                            float* __restrict__ Rbf, float* __restrict__ Vn,
                            float* __restrict__ Cut) {
    int e = blockIdx.x * blockDim.x + threadIdx.x;
    if (e >= E) return;
    int s = ei[e], d = ei[E + e];
    float vx = pos[s * 3 + 0] - pos[d * 3 + 0];
    float vy = pos[s * 3 + 1] - pos[d * 3 + 1];
    float vz = pos[s * 3 + 2] - pos[d * 3 + 2];
    float w = sqrtf(vx * vx + vy * vy + vz * vz);
    float denom = (s == d) ? 1.0f : w;
    Vn[e * 3 + 0] = vx / denom;
    Vn[e * 3 + 1] = vy / denom;
    Vn[e * 3 + 2] = vz / denom;
    float cut = (w < CUTOFF) ? 0.5f * (cosf(w * (3.14159265358979f / CUTOFF)) + 1.0f) : 0.0f;
    Cut[e] = cut;
    const float alpha = 5.0f / CUTOFF;
    const float start = expf(-CUTOFF);
    float tb = (2.0f / 32.0f) * (1.0f - start);
    float beta = 1.0f / (tb * tb);
    float ew = expf(-alpha * w);
    #pragma unroll
    for (int i = 0; i < 32; i++) {
        float mean = start + (1.0f - start) * ((float)i / 31.0f);
        float t = ew - mean;
        Rbf[(size_t)e * 32 + i] = cut * expf(-beta * t * t);
    }
}

// Zcat[e, 0:128] = Zemb[src[e]], Zcat[e, 128:256] = Zemb[dst[e]]
__global__ void k_zcat(const float* __restrict__ Zemb, const int* __restrict__ ei, int E,
                       float* __restrict__ Zcat) {
    size_t i = (size_t)blockIdx.x * blockDim.x + threadIdx.x;
    if (i >= (size_t)E * 256) return;
    int e = (int)(i >> 8), k = (int)(i & 255);
    int node = (k < HD) ? ei[e] : ei[E + e];
    Zcat[i] = Zemb[(size_t)node * HD + (k & 127)];
}

// ------------------------------------------------------- generic f32 WMMA GEMM
// out[m,n] = act( ((A[m,:] . W[n,:]) + bias[n]) * rowscale[m] )
// block = 256 thr (8 waves); block tile = 16 rows x 128 cols; A tile in LDS.
__global__ void k_gemm_wmma(const float* __restrict__ A, const float* __restrict__ W,
                            const float* __restrict__ bias, const float* __restrict__ rowscale,
                            float* __restrict__ out, int M, int N, int K, int act) {
    extern __shared__ float Alds[];
    int m0 = blockIdx.y * 16;
    int tid = threadIdx.x;
    for (int i = tid; i < 16 * K; i += 256) {
        int r = i / K, c = i - r * K;
        Alds[i] = A[(size_t)(m0 + r) * K + c];
    }
    __syncthreads();
    int wave = tid >> 5, lane = tid & 31, half = lane >> 4, li = lane & 15;
    int n0 = blockIdx.x * 128 + wave * 16;
    if (n0 >= N) return;                 // wave-uniform; EXEC stays all-ones for WMMA
    int n = n0 + li;
    const float* Wrow = W + (size_t)n * K;
    const float* Arow = Alds + li * K;
    v8f acc = {};
    for (int k = 0; k < K; k += 4) {
        v2f a, b;
        a[0] = Arow[k + 2 * half];
        a[1] = Arow[k + 2 * half + 1];
        b[0] = Wrow[k + 2 * half];
        b[1] = Wrow[k + 2 * half + 1];
        acc = __builtin_amdgcn_wmma_f32_16x16x4_f32(false, a, false, b, (short)0, acc,
                                                    false, false);
    }
    float bv = bias ? bias[n] : 0.0f;
    #pragma unroll
    for (int r = 0; r < 8; r++) {
        int m = m0 + r + half * 8;
        float v = acc[r] + bv;
        if (rowscale) v *= rowscale[m];
        if (act == 1) v = v / (1.0f + expf(-v));   // SiLU
        out[(size_t)m * N + n] = v;
    }
}

// -------------------------------- fused rbf-projection (WMMA) + tensor scatter
// D[e, 0:384] = rbf[e,:32] @ [dp1;dp2;dp3]^T ; c = (D+b)*Cmat ; atomic scatter
// into 10 accumulator planes (i0 | a_xyz | sym6), layout (10, N, 128).
__global__ void k_dproj_scatter(const float* __restrict__ Rbf, const float* __restrict__ Cmat,
                                const float* __restrict__ Vn, const int* __restrict__ srcIdx,
                                const float* __restrict__ w1, const float* __restrict__ w2,
                                const float* __restrict__ w3, const float* __restrict__ b1,
                                const float* __restrict__ b2, const float* __restrict__ b3,
                                float* __restrict__ Acc, int Nn) {
    __shared__ float Alds[16 * 32];
    int e0 = blockIdx.x * 16;
    int tid = threadIdx.x;
    for (int i = tid; i < 16 * 32; i += 256) Alds[i] = Rbf[(size_t)e0 * 32 + i];
    __syncthreads();
    int wave = tid >> 5, lane = tid & 31, half = lane >> 4, li = lane & 15;
    size_t PS = (size_t)Nn * HD;
    for (int t = wave; t < 24; t += 8) {          // 24 col tiles of 16 over 384 outputs
        int n = t * 16 + li;
        int comp = n >> 7, h = n & 127;
        const float* W = (comp == 0) ? w1 : (comp == 1 ? w2 : w3);
        const float* B = (comp == 0) ? b1 : (comp == 1 ? b2 : b3);
        const float* Wrow = W + h * 32;
        const float* Arow = Alds + li * 32;
        v8f acc = {};
        #pragma unroll
        for (int k = 0; k < 32; k += 4) {
            v2f a, b;
            a[0] = Arow[k + 2 * half];
            a[1] = Arow[k + 2 * half + 1];
            b[0] = Wrow[k + 2 * half];
            b[1] = Wrow[k + 2 * half + 1];
            acc = __builtin_amdgcn_wmma_f32_16x16x4_f32(false, a, false, b, (short)0, acc,
                                                        false, false);
        }
        float bv = B[h];
        #pragma unroll
        for (int r = 0; r < 8; r++) {
            int e = e0 + r + half * 8;
            float val = (acc[r] + bv) * Cmat[(size_t)e * HD + h];
            int node = srcIdx[e];
            float* base = Acc + (size_t)node * HD + h;
            if (comp == 0) {
                atomicAdd(base, val);                       // I: scalar * eye
            } else if (comp == 1) {
                float vx = Vn[e * 3], vy = Vn[e * 3 + 1], vz = Vn[e * 3 + 2];
                atomicAdd(base + 1 * PS, val * vx);         // A: skew(vector)
                atomicAdd(base + 2 * PS, val * vy);
                atomicAdd(base + 3 * PS, val * vz);
            } else {
                float vx = Vn[e * 3], vy = Vn[e * 3 + 1], vz = Vn[e * 3 + 2];
                float tr3 = (vx * vx + vy * vy + vz * vz) * (1.0f / 3.0f);
                atomicAdd(base + 4 * PS, val * (vx * vx - tr3));  // S: sym traceless
                atomicAdd(base + 5 * PS, val * (vx * vy));
                atomicAdd(base + 6 * PS, val * (vx * vz));
                atomicAdd(base + 7 * PS, val * (vy * vy - tr3));
                atomicAdd(base + 8 * PS, val * (vy * vz));
                atomicAdd(base + 9 * PS, val * (vz * vz - tr3));
            }
        }
    }
}

// ||I0+A0+S0||^2 per (n,h) then layernorm over H (one block per node)
__global__ void k_tnorm_ln(const float* __restrict__ Acc, const float* __restrict__ g,
                           const float* __restrict__ b, float* __restrict__ out, int Nn) {
    int n = blockIdx.x, h = threadIdx.x;
    size_t PS = (size_t)Nn * HD;
    size_t idx = (size_t)n * HD + h;
    float i0 = Acc[idx];
    float ax = Acc[idx + 1 * PS], ay = Acc[idx + 2 * PS], az = Acc[idx + 3 * PS];
    float sxx = Acc[idx + 4 * PS], sxy = Acc[idx + 5 * PS], sxz = Acc[idx + 6 * PS];
    float syy = Acc[idx + 7 * PS], syz = Acc[idx + 8 * PS], szz = Acc[idx + 9 * PS];
    float v = (i0 + sxx) * (i0 + sxx) + (i0 + syy) * (i0 + syy) + (i0 + szz) * (i0 + szz)
            + 2.0f * (sxy * sxy + az * az) + 2.0f * (sxz * sxz + ay * ay)
            + 2.0f * (syz * syz + ax * ax);
    __shared__ float s1[HD], s2[HD];
    s1[h] = v; s2[h] = v * v;
    __syncthreads();
    for (int st = 64; st > 0; st >>= 1) {
        if (h < st) { s1[h] += s1[h + st]; s2[h] += s2[h + st]; }
        __syncthreads();
    }
    float m = s1[0] * (1.0f / HD);
    float var = s2[0] * (1.0f / HD) - m * m;
    out[idx] = (v - m) * rsqrtf(var + 1e-5f) * g[h] + b[h];
}

// closed-form tensor norms of transformed I/A/S scaled by nrm, then layernorm(384)
__global__ void k_xnorm_ln(const float* __restrict__ Acc2, const float* __restrict__ Nrm,
                           const float* __restrict__ g, const float* __restrict__ b,
                           float* __restrict__ X, int Nn) {
    int n = blockIdx.x, t = threadIdx.x;
    int k = t & 127, sec = t >> 7;
    size_t PS = (size_t)Nn * HD;
    size_t idx = (size_t)n * HD + k;
    float nr = Nrm[(size_t)n * 384 + k * 3 + sec];
    float v;
    if (sec == 0) {
        float i0 = Acc2[idx];
        v = 3.0f * i0 * i0 * nr * nr;
    } else if (sec == 1) {
        float ax = Acc2[idx + 1 * PS], ay = Acc2[idx + 2 * PS], az = Acc2[idx + 3 * PS];
        v = 2.0f * (ax * ax + ay * ay + az * az) * nr * nr;
    } else {
        float sxx = Acc2[idx + 4 * PS], sxy = Acc2[idx + 5 * PS], sxz = Acc2[idx + 6 * PS];
        float syy = Acc2[idx + 7 * PS], syz = Acc2[idx + 8 * PS], szz = Acc2[idx + 9 * PS];
        v = (sxx * sxx + syy * syy + szz * szz
           + 2.0f * (sxy * sxy + sxz * sxz + syz * syz)) * nr * nr;
    }
    __shared__ float sa[384], sb1[128], sa2[384], sb2[128];
    sa[t] = v; sa2[t] = v * v;
    __syncthreads();
    if (t < 128) {
        sb1[t] = sa[t] + sa[t + 128] + sa[t + 256];
        sb2[t] = sa2[t] + sa2[t + 128] + sa2[t + 256];
    }
    __syncthreads();
    for (int st = 64; st > 0; st >>= 1) {
        if (t < st) { sb1[t] += sb1[t + st]; sb2[t] += sb2[t + st]; }
        __syncthreads();
    }
    float m = sb1[0] * (1.0f / 384.0f);
    float var = sb2[0] * (1.0f / 384.0f) - m * m;
    int col = sec * HD + k;
    X[(size_t)n * 384 + col] = (v - m) * rsqrtf(var + 1e-5f) * g[col] + b[col];
}

// y[n] = H3[n,:64] . ol2_w + ol2_b
__global__ void k_final(const float* __restrict__ H3, const float* __restrict__ w,
                        const float* __restrict__ bb, float* __restrict__ y, int Nn) {
    int n = blockIdx.x * blockDim.x + threadIdx.x;
    if (n >= Nn) return;
    float s = bb[0];
    #pragma unroll
    for (int k = 0; k < 64; k++) s += H3[(size_t)n * 64 + k] * w[k];
    y[n] = s;
}

// ---------------------------------------------------------------------- host
extern "C" void kernel_launch(void* const* d_in, const int* in_sizes, int n_in,
                              void* d_out, int out_size, void* d_ws, size_t ws_size,
                              hipStream_t stream) {
    const int*   z       = (const int*)d_in[0];
    const float* pos     = (const float*)d_in[1];
    const int*   ei      = (const int*)d_in[3];   // [2,E] (src rows then dst rows)
    const float* emb_w   = (const float*)d_in[4];
    const float* emb2_w  = (const float*)d_in[5];
    const float* emb2_b  = (const float*)d_in[6];
    const float* dp1_w   = (const float*)d_in[7];
    const float* dp1_b   = (const float*)d_in[8];
    const float* dp2_w   = (const float*)d_in[9];
    const float* dp2_b   = (const float*)d_in[10];
    const float* dp3_w   = (const float*)d_in[11];
    const float* dp3_b   = (const float*)d_in[12];
    const float* lt0_w   = (const float*)d_in[13];
    const float* lt1_w   = (const float*)d_in[14];
    const float* lt2_w   = (const float*)d_in[15];
    const float* ls0_w   = (const float*)d_in[16];
    const float* ls0_b   = (const float*)d_in[17];
    const float* ls1_w   = (const float*)d_in[18];
    const float* ls1_b   = (const float*)d_in[19];
    const float* in_g    = (const float*)d_in[20];
    const float* in_b    = (const float*)d_in[21];
    const float* lin_w   = (const float*)d_in[22];
    const float* lin_b   = (const float*)d_in[23];
    const float* on_g    = (const float*)d_in[24];
    const float* on_b    = (const float*)d_in[25];
    const float* ol1_w   = (const float*)d_in[26];
    const float* ol1_b   = (const float*)d_in[27];
    const float* ol2_w   = (const float*)d_in[28];
    const float* ol2_b   = (const float*)d_in[29];
    int Nn = in_sizes[0];
    int E  = in_sizes[3] / 2;
    float* y = (float*)d_out;

    size_t off = 0;
    auto alloc = [&](size_t nel) {
        float* p = (float*)((char*)d_ws + off);
        off += nel * sizeof(float);
        return p;
    };
    float* Zemb = alloc((size_t)Nn * HD);
    float* Zcat = alloc((size_t)E * 256);
    float* Cmat = alloc((size_t)E * HD);
    float* Rbf  = alloc((size_t)E * 32);
    float* Vn   = alloc((size_t)E * 3);
    float* Cut  = alloc((size_t)E);
    float* Acc  = alloc((size_t)10 * Nn * HD);
    float* Tn   = alloc((size_t)Nn * HD);
    float* H1   = alloc((size_t)Nn * 256);
    float* Nrm  = alloc((size_t)Nn * 384);
    float* Acc2 = alloc((size_t)10 * Nn * HD);
    float* X    = alloc((size_t)Nn * 384);
    float* H2   = alloc((size_t)Nn * HD);
    float* H3   = alloc((size_t)Nn * 64);

    size_t accN = (size_t)10 * Nn * HD;
    k_zero<<<dim3((unsigned)((accN + 255) / 256)), 256, 0, stream>>>(Acc, accN);
    k_embed<<<dim3((unsigned)(((size_t)Nn * HD + 255) / 256)), 256, 0, stream>>>(Zemb, emb_w, z, Nn);
    k_edge_geom<<<dim3((E + 127) / 128), 128, 0, stream>>>(pos, ei, E, Rbf, Vn, Cut);
    k_zcat<<<dim3((unsigned)(((size_t)E * 256 + 255) / 256)), 256, 0, stream>>>(Zemb, ei, E, Zcat);

    // C = cutoff * (Zcat @ emb2_w^T + b)       -- (E,256)x(256,128)
    k_gemm_wmma<<<dim3(1, E / 16), 256, 16 * 256 * 4, stream>>>(
        Zcat, emb2_w, emb2_b, Cut, Cmat, E, 128, 256, 0);

    // rbf projections (32->384) fused with tensor scatter
    k_dproj_scatter<<<dim3(E / 16), 256, 0, stream>>>(
        Rbf, Cmat, Vn, ei, dp1_w, dp2_w, dp3_w, dp1_b, dp2_b, dp3_b, Acc, Nn);

    k_tnorm_ln<<<dim3(Nn), 128, 0, stream>>>(Acc, in_g, in_b, Tn, Nn);

    // nrm MLP: 128->256 silu, 256->384 silu
    k_gemm_wmma<<<dim3(2, Nn / 16), 256, 16 * 128 * 4, stream>>>(
        Tn, ls0_w, ls0_b, nullptr, H1, Nn, 256, 128, 1);
    k_gemm_wmma<<<dim3(3, Nn / 16), 256, 16 * 256 * 4, stream>>>(
        H1, ls1_w, ls1_b, nullptr, Nrm, Nn, 384, 256, 1);

    // 10 channel-mixing GEMMs (lt0 for i0, lt1 for vector, lt2 for sym6)
    for (int p = 0; p < 10; p++) {
        const float* Wp = (p == 0) ? lt0_w : (p <= 3 ? lt1_w : lt2_w);
        k_gemm_wmma<<<dim3(1, Nn / 16), 256, 16 * 128 * 4, stream>>>(
            Acc + (size_t)p * Nn * HD, Wp, nullptr, nullptr,
            Acc2 + (size_t)p * Nn * HD, Nn, 128, 128, 0);
    }

    k_xnorm_ln<<<dim3(Nn), 384, 0, stream>>>(Acc2, Nrm, on_g, on_b, X, Nn);

    // head: 384->128 silu, 128->64 silu, 64->1
    k_gemm_wmma<<<dim3(1, Nn / 16), 256, 16 * 384 * 4, stream>>>(
        X, lin_w, lin_b, nullptr, H2, Nn, 128, 384, 1);
    k_gemm_wmma<<<dim3(1, Nn / 16), 256, 16 * 128 * 4, stream>>>(
        H2, ol1_w, ol1_b, nullptr, H3, Nn, 64, 128, 1);
    k_final<<<dim3((Nn + 255) / 256), 256, 0, stream>>>(H3, ol2_w, ol2_b, y, Nn);
}